// DocLSTM_7078106104363
// MI455X (gfx1250) — compile-verified
//
#include <hip/hip_runtime.h>

// ---------------------------------------------------------------------------
// CDNA5 (gfx1250) fp32 WMMA: V_WMMA_F32_16X16X4_F32
//   A: 16x4  -> v2f/lane (lanes 0-15: K=0,1 ; lanes 16-31: K=2,3)
//   B:  4x16 -> v2f/lane (same K split, N = lane&15)
//   C/D: 16x16 -> v8f/lane (VGPR v: M = v + 8*(lane>=16), N = lane&15)
// ---------------------------------------------------------------------------
typedef float v2f __attribute__((ext_vector_type(2)));
typedef float v8f __attribute__((ext_vector_type(8)));

__device__ __forceinline__ v8f wmma4(v2f a, v2f b, v8f c) {
  return __builtin_amdgcn_wmma_f32_16x16x4_f32(
      false, a, false, b, (short)0, c, false, false);
}

__device__ __forceinline__ float sigm(float x) { return 1.0f / (1.0f + expf(-x)); }

#define MEMD   300
#define GATES3 900
#define GATES4 1200
#define NSENT  128
#define LTOK   32
#define NTOK   4096

// ---------------------------------------------------------------------------
// k_b2: b2[o] = dot(iouh_w[o,:], Wa_b)
// ---------------------------------------------------------------------------
__global__ void k_b2(const float* __restrict__ iouh_w,
                     const float* __restrict__ wa_b,
                     float* __restrict__ b2) {
  int o = blockIdx.x * blockDim.x + threadIdx.x;
  if (o >= GATES3) return;
  const float* w = iouh_w + o * MEMD;
  float acc = 0.0f;
  for (int j = 0; j < MEMD; ++j) acc += w[j] * wa_b[j];
  b2[o] = acc;
}

// ---------------------------------------------------------------------------
// k_m2: Wcomb[0:900,:] = iouh_w(900x300) @ Wa_w(300x300)  (WMMA, B is KxN)
// ---------------------------------------------------------------------------
__global__ void k_m2(const float* __restrict__ iouh_w,
                     const float* __restrict__ wa_w,
                     float* __restrict__ Wcomb) {
  const int lane = threadIdx.x & 31;
  const int wave = threadIdx.x >> 5;
  int tile = blockIdx.x * (blockDim.x >> 5) + wave;
  if (tile >= 57 * 19) return;                 // wave-uniform exit
  const int mt = tile / 19, nt = tile % 19;
  const int mBase = mt * 16, nBase = nt * 16;
  const int fm = lane & 15;
  const int k2 = (lane >> 4) << 1;

  int arow = mBase + fm; if (arow > 899) arow = 899;
  const float* aRow = iouh_w + arow * MEMD;
  int ncol = nBase + fm; if (ncol > 299) ncol = 299;

  v8f acc = {};
  for (int kk = 0; kk < MEMD; kk += 4) {
    v2f a, b;
    a.x = aRow[kk + k2];
    a.y = aRow[kk + k2 + 1];
    b.x = wa_w[(kk + k2) * MEMD + ncol];
    b.y = wa_w[(kk + k2 + 1) * MEMD + ncol];
    acc = wmma4(a, b, acc);
  }
  const int mOff = (lane >> 4) * 8;
  if (nBase + fm < MEMD) {
    for (int v = 0; v < 8; ++v) {
      int row = mBase + mOff + v;
      if (row < GATES3) Wcomb[row * MEMD + nBase + fm] = acc[v];
    }
  }
}

// ---------------------------------------------------------------------------
// k_xpre: Xpre[t*128+sent, :] = emb[tok] @ [ioux_w; fx_w]^T + fused biases
// 2-way M-blocked WMMA GEMM: each B (weight) fragment feeds two WMMAs.
// tiles: 128 M-pairs x 75 N  = 9600 waves
// ---------------------------------------------------------------------------
__global__ void k_xpre(const int* __restrict__ body,
                       const float* __restrict__ emb,
                       const float* __restrict__ ioux_w,
                       const float* __restrict__ fx_w,
                       const float* __restrict__ ioux_b,
                       const float* __restrict__ iouh_b,
                       const float* __restrict__ fx_b,
                       const float* __restrict__ fh_b,
                       const float* __restrict__ b2,
                       float* __restrict__ Xpre) {
  const int lane = threadIdx.x & 31;
  const int wave = threadIdx.x >> 5;
  const int NT = GATES4 / 16;                  // 75
  int tile = blockIdx.x * (blockDim.x >> 5) + wave;
  if (tile >= (NTOK / 32) * NT) return;        // wave-uniform exit
  const int mt = tile / NT, nt = tile % NT;
  const int mBase = mt * 32, nBase = nt * 16;
  const int fm = lane & 15;
  const int k2 = (lane >> 4) << 1;

  // token gather for both M-tiles (t-major row layout: r = t*128 + sent)
  const int r0 = mBase + fm;
  const int r1 = r0 + 16;
  const int t0 = r0 >> 7, s0 = r0 & 127;
  const int t1 = r1 >> 7, s1 = r1 & 127;
  const float* aRow0 = emb + body[s0 * LTOK + (LTOK - 1 - t0)] * MEMD;
  const float* aRow1 = emb + body[s1 * LTOK + (LTOK - 1 - t1)] * MEMD;

  const int ncol = nBase + fm;
  const float* bRow = (ncol < GATES3) ? (ioux_w + ncol * MEMD)
                                      : (fx_w + (ncol - GATES3) * MEMD);
  v8f acc0 = {}, acc1 = {};
  for (int kk = 0; kk < MEMD; kk += 4) {
    v2f a0, a1, b;
    b.x  = bRow[kk + k2];  b.y  = bRow[kk + k2 + 1];
    a0.x = aRow0[kk + k2]; a0.y = aRow0[kk + k2 + 1];
    a1.x = aRow1[kk + k2]; a1.y = aRow1[kk + k2 + 1];
    acc0 = wmma4(a0, b, acc0);
    acc1 = wmma4(a1, b, acc1);
  }
  const float bias = (ncol < GATES3)
                         ? (ioux_b[ncol] + iouh_b[ncol] + b2[ncol])
                         : (fx_b[ncol - GATES3] + fh_b[ncol - GATES3]);
  const int mOff = (lane >> 4) * 8;
  for (int v = 0; v < 8; ++v) {
    int row = mBase + mOff + v;
    Xpre[row * GATES4 + ncol]        = acc0[v] + bias;
    Xpre[(row + 16) * GATES4 + ncol] = acc1[v] + bias;
  }
}

// ---------------------------------------------------------------------------
// k_tree: FUSED 32-step tree-LSTM. One WG = one 16-sentence M-tile (8 WGs).
// C,H state and the 16x1200 gate matrix live in LDS (115.2 KB dynamic).
// Per step: G = Xpre_t + C @ Wcomb^T via WMMA (A fragments from LDS),
// then gate nonlinearities update C,H in LDS. Final C -> sentc_out.
// ---------------------------------------------------------------------------
__global__ void __launch_bounds__(256)
k_tree(const float* __restrict__ Wcomb,
       const float* __restrict__ Xpre,
       float* __restrict__ sentc_out) {
  extern __shared__ float smem[];
  float* Csh = smem;                // 16*300
  float* Hsh = smem + 4800;         // 16*300
  float* Gsh = smem + 9600;         // 16*1200
  const int tid  = threadIdx.x;
  const int lane = tid & 31;
  const int wave = tid >> 5;
  const int fm   = lane & 15;
  const int k2   = (lane >> 4) << 1;
  const int mOff = (lane >> 4) * 8;
  const int wgSent0 = blockIdx.x * 16;

  for (int i = tid; i < 4800; i += 256) { Csh[i] = 0.0f; Hsh[i] = 0.0f; }
  __syncthreads();

  for (int t = 0; t < LTOK; ++t) {
    const float* Xt = Xpre + (t * NSENT + wgSent0) * GATES4;
    // G = Xt + Csh(16x300) @ Wcomb^T(1200x300)
    for (int nt = wave; nt < GATES4 / 16; nt += 8) {
      const int ncol = nt * 16 + fm;
      const float* bRow = Wcomb + ncol * MEMD;
      v8f acc = {};
      for (int kk = 0; kk < MEMD; kk += 4) {
        v2f a, b;
        a.x = Csh[fm * MEMD + kk + k2];
        a.y = Csh[fm * MEMD + kk + k2 + 1];
        b.x = bRow[kk + k2];
        b.y = bRow[kk + k2 + 1];
        acc = wmma4(a, b, acc);
      }
      for (int v = 0; v < 8; ++v) {
        int row = mOff + v;
        Gsh[row * GATES4 + ncol] = acc[v] + Xt[row * GATES4 + ncol];
      }
    }
    __syncthreads();
    // gates: ch_h = c_prev feeds gates; ch_c = h_prev feeds f*ch_c (per ref)
    for (int idx = tid; idx < 16 * MEMD; idx += 256) {
      const int m = idx / MEMD, j = idx % MEMD;
      const float* g = Gsh + m * GATES4;
      const float gi = g[j], go = g[MEMD + j], gu = g[2 * MEMD + j],
                  gf = g[3 * MEMD + j];
      const float cn = sigm(gi) * tanhf(gu) + sigm(gf) * Hsh[idx];
      const float hn = sigm(go) * tanhf(cn);
      Csh[idx] = cn;
      Hsh[idx] = hn;
    }
    __syncthreads();
  }
  for (int idx = tid; idx < 16 * MEMD; idx += 256)
    sentc_out[wgSent0 * MEMD + idx] = Csh[idx];
}

// ---------------------------------------------------------------------------
// k_para: FUSED 16-step para-LSTM, 8 paragraphs batched into one 16-row
// WMMA M-tile (rows 8-15 clamped/zero). Single WG; h,c,g in LDS.
// Writes para_h (8x300) to paraH_out.
// ---------------------------------------------------------------------------
__global__ void __launch_bounds__(256)
k_para(const float* __restrict__ sentc,
       const float* __restrict__ Wih, const float* __restrict__ Whh,
       const float* __restrict__ bih, const float* __restrict__ bhh,
       float* __restrict__ paraH_out) {
  extern __shared__ float smem[];
  float* hPs = smem;                // 16*300 (rows 8..15 stay zero)
  float* cPs = smem + 4800;         // 16*300
  float* gPs = smem + 9600;         // 16*1200
  const int tid  = threadIdx.x;
  const int lane = tid & 31;
  const int wave = tid >> 5;
  const int fm   = lane & 15;
  const int k2   = (lane >> 4) << 1;
  const int mOff = (lane >> 4) * 8;

  for (int i = tid; i < 4800; i += 256) { hPs[i] = 0.0f; cPs[i] = 0.0f; }
  __syncthreads();

  const int pcl = (fm > 7) ? 7 : fm;           // clamp padded rows
  for (int s = 0; s < 16; ++s) {
    const float* ax = sentc + (pcl * 16 + s) * MEMD;   // sent_enc[p][s]
    for (int nt = wave; nt < GATES4 / 16; nt += 8) {
      const int ncol = nt * 16 + fm;
      const float* bx = Wih + ncol * MEMD;
      const float* bh = Whh + ncol * MEMD;
      v8f acc = {};
      for (int kk = 0; kk < MEMD; kk += 4) {     // x @ Wih^T
        v2f a, b;
        a.x = ax[kk + k2]; a.y = ax[kk + k2 + 1];
        b.x = bx[kk + k2]; b.y = bx[kk + k2 + 1];
        acc = wmma4(a, b, acc);
      }
      for (int kk = 0; kk < MEMD; kk += 4) {     // h @ Whh^T (A from LDS)
        v2f a, b;
        a.x = hPs[fm * MEMD + kk + k2];
        a.y = hPs[fm * MEMD + kk + k2 + 1];
        b.x = bh[kk + k2]; b.y = bh[kk + k2 + 1];
        acc = wmma4(a, b, acc);
      }
      const float bias = bih[ncol] + bhh[ncol];
      for (int v = 0; v < 8; ++v)
        gPs[(mOff + v) * GATES4 + ncol] = acc[v] + bias;
    }
    __syncthreads();
    for (int idx = tid; idx < 8 * MEMD; idx += 256) {   // i,f,g,o split
      const int m = idx / MEMD, j = idx % MEMD;
      const float* g = gPs + m * GATES4;
      const float gi = g[j], gf = g[MEMD + j], gg = g[2 * MEMD + j],
                  go = g[3 * MEMD + j];
      const float cn = sigm(gf) * cPs[idx] + sigm(gi) * tanhf(gg);
      const float hn = sigm(go) * tanhf(cn);
      cPs[idx] = cn;
      hPs[idx] = hn;
    }
    __syncthreads();
  }
  for (int idx = tid; idx < 8 * MEMD; idx += 256) paraH_out[idx] = hPs[idx];
}

// ---------------------------------------------------------------------------
// k_body: 8-step LSTM over para_h, single workgroup -> out[0:300]
// ---------------------------------------------------------------------------
__global__ void __launch_bounds__(1024)
k_body(const float* __restrict__ paraH,
       const float* __restrict__ Wih, const float* __restrict__ Whh,
       const float* __restrict__ bih, const float* __restrict__ bhh,
       float* __restrict__ out) {
  __shared__ float x[MEMD], h[MEMD], c[MEMD], g[GATES4];
  const int tid = threadIdx.x;
  for (int j = tid; j < MEMD; j += blockDim.x) { h[j] = 0.0f; c[j] = 0.0f; }
  __syncthreads();
  for (int p = 0; p < 8; ++p) {
    for (int j = tid; j < MEMD; j += blockDim.x) x[j] = paraH[p * MEMD + j];
    __syncthreads();
    for (int n = tid; n < GATES4; n += blockDim.x) {
      const float* wi = Wih + n * MEMD;
      const float* wh = Whh + n * MEMD;
      float acc = bih[n] + bhh[n];
      for (int k = 0; k < MEMD; ++k) acc += x[k] * wi[k] + h[k] * wh[k];
      g[n] = acc;
    }
    __syncthreads();
    for (int j = tid; j < MEMD; j += blockDim.x) {
      const float gi = g[j], gf = g[MEMD + j], gg = g[2 * MEMD + j],
                  go = g[3 * MEMD + j];
      const float cn = sigm(gf) * c[j] + sigm(gi) * tanhf(gg);
      const float hn = sigm(go) * tanhf(cn);
      c[j] = cn; h[j] = hn;
    }
    __syncthreads();
  }
  for (int j = tid; j < MEMD; j += blockDim.x) out[j] = h[j];
}

// ---------------------------------------------------------------------------
// k_rsent: 24-step tree-LSTM (indicator 0, chs = c_prev) -> out[300:600]
// ---------------------------------------------------------------------------
__global__ void __launch_bounds__(1024)
k_rsent(const int* __restrict__ rsent, const float* __restrict__ emb,
        const float* __restrict__ ioux_w, const float* __restrict__ ioux_b,
        const float* __restrict__ iouh_w, const float* __restrict__ iouh_b,
        const float* __restrict__ fx_w, const float* __restrict__ fx_b,
        const float* __restrict__ fh_w, const float* __restrict__ fh_b,
        float* __restrict__ out) {
  __shared__ float x[MEMD], c[MEMD], h[MEMD], g[GATES4];
  const int tid = threadIdx.x;
  for (int j = tid; j < MEMD; j += blockDim.x) { c[j] = 0.0f; h[j] = 0.0f; }
  __syncthreads();
  for (int t = 0; t < 24; ++t) {
    const int tok = rsent[23 - t];
    for (int j = tid; j < MEMD; j += blockDim.x) x[j] = emb[tok * MEMD + j];
    __syncthreads();
    for (int n = tid; n < GATES4; n += blockDim.x) {
      float acc;
      const float* wx;
      const float* wc;
      if (n < GATES3) {
        acc = ioux_b[n] + iouh_b[n];
        wx = ioux_w + n * MEMD;
        wc = iouh_w + n * MEMD;       // chs = ch_h = c_prev
      } else {
        const int q = n - GATES3;
        acc = fx_b[q] + fh_b[q];
        wx = fx_w + q * MEMD;
        wc = fh_w + q * MEMD;
      }
      for (int k = 0; k < MEMD; ++k) acc += x[k] * wx[k] + c[k] * wc[k];
      g[n] = acc;
    }
    __syncthreads();
    for (int j = tid; j < MEMD; j += blockDim.x) {
      const float gi = g[j], go = g[MEMD + j], gu = g[2 * MEMD + j],
                  gf = g[3 * MEMD + j];
      const float cn = sigm(gi) * tanhf(gu) + sigm(gf) * h[j];  // ch_c = h_prev
      const float hn = sigm(go) * tanhf(cn);
      c[j] = cn; h[j] = hn;
    }
    __syncthreads();
  }
  for (int j = tid; j < MEMD; j += blockDim.x) out[MEMD + j] = c[j];
}

// ---------------------------------------------------------------------------
// Host-side orchestration: 8 graph nodes total.
// ---------------------------------------------------------------------------
extern "C" void kernel_launch(void* const* d_in, const int* in_sizes, int n_in,
                              void* d_out, int out_size, void* d_ws, size_t ws_size,
                              hipStream_t stream) {
  (void)in_sizes; (void)n_in; (void)out_size; (void)ws_size;
  const int*   rsent   = (const int*)d_in[0];
  const int*   body    = (const int*)d_in[1];
  const float* emb     = (const float*)d_in[2];
  const float* ioux_w  = (const float*)d_in[3];
  const float* ioux_b  = (const float*)d_in[4];
  const float* iouh_w  = (const float*)d_in[5];
  const float* iouh_b  = (const float*)d_in[6];
  const float* fx_w    = (const float*)d_in[7];
  const float* fx_b    = (const float*)d_in[8];
  const float* fh_w    = (const float*)d_in[9];
  const float* fh_b    = (const float*)d_in[10];
  const float* wa_w    = (const float*)d_in[11];
  const float* wa_b    = (const float*)d_in[12];
  const float* pWih    = (const float*)d_in[13];
  const float* pWhh    = (const float*)d_in[14];
  const float* pbih    = (const float*)d_in[15];
  const float* pbhh    = (const float*)d_in[16];
  const float* bWih    = (const float*)d_in[17];
  const float* bWhh    = (const float*)d_in[18];
  const float* bbih    = (const float*)d_in[19];
  const float* bbhh    = (const float*)d_in[20];
  float* out = (float*)d_out;

  // Workspace layout (floats): ~21.3 MB
  float* ws    = (float*)d_ws;
  float* Xpre  = ws;                         // 4096*1200
  float* Wcomb = Xpre + NTOK * GATES4;       // 1200*300
  float* b2    = Wcomb + GATES4 * MEMD;      // 1024 (900 used)
  float* sentc = b2 + 1024;                  // 128*300
  float* paraH = sentc + NSENT * MEMD;       // 8*300

  const size_t fusedLds = (size_t)(4800 + 4800 + 19200) * sizeof(float); // 115.2KB

  // 1) b2 = iouh_w @ Wa_b
  k_b2<<<dim3(4), dim3(256), 0, stream>>>(iouh_w, wa_b, b2);
  // 2) Wcomb[0:900] = iouh_w @ Wa_w
  k_m2<<<dim3(271), dim3(128), 0, stream>>>(iouh_w, wa_w, Wcomb);
  // 3) Wcomb[900:1200] = fh_w
  hipMemcpyAsync(Wcomb + GATES3 * MEMD, fh_w, MEMD * MEMD * sizeof(float),
                 hipMemcpyDeviceToDevice, stream);
  // 4) Xpre: gathered-embedding projection GEMM (2-way M-blocked WMMA)
  k_xpre<<<dim3(1200), dim3(256), 0, stream>>>(body, emb, ioux_w, fx_w, ioux_b,
                                               iouh_b, fx_b, fh_b, b2, Xpre);
  // 5) fused 32-step tree-LSTM: 8 WGs x 16 sentences, state in LDS
  k_tree<<<dim3(8), dim3(256), fusedLds, stream>>>(Wcomb, Xpre, sentc);
  // 6) fused 16-step para-LSTM: 1 WG, 8 paragraphs in one WMMA tile
  k_para<<<dim3(1), dim3(256), fusedLds, stream>>>(sentc, pWih, pWhh,
                                                   pbih, pbhh, paraH);
  // 7) body LSTM -> out[0:300]
  k_body<<<dim3(1), dim3(1024), 0, stream>>>(paraH, bWih, bWhh, bbih, bbhh, out);
  // 8) rsent tree-LSTM -> out[300:600]
  k_rsent<<<dim3(1), dim3(1024), 0, stream>>>(rsent, emb, ioux_w, ioux_b,
                                              iouh_w, iouh_b, fx_w, fx_b,
                                              fh_w, fh_b, out);
}